// SparseMoELoRA_75531294868086
// MI455X (gfx1250) — compile-verified
//
#include <hip/hip_runtime.h>

// Problem constants (from reference):
//   B=4, L=2048 -> N_TOK=8192 tokens; C=4096; E=8 experts; K=2 top-k;
//   R=64 rank; O=4096; SCALE = 64/64 = 1
#define N_TOK 8192
#define C_DIM 4096
#define E_NUM 8
#define R_RANK 64
#define O_DIM 4096
#define JT (E_NUM * R_RANK)       /* 512 = concatenated low-rank width */
#define LORA_SCALE (64.0f / 64.0f)

typedef __attribute__((ext_vector_type(2))) float v2f;
typedef __attribute__((ext_vector_type(8))) float v8f;

// ---------------------------------------------------------------------------
// Kernel 1: noisy top-k router. One wave32 per token.
// Each lane strides C with float4 loads, accumulating 8 Wr-dots + 8 Wn-dots.
// Butterfly reduce across the wave, then compute noisy logits, top-2, and the
// 2-way softmax gates; store dense gate table g[N_TOK, E] (zeros elsewhere).
// ---------------------------------------------------------------------------
__global__ __launch_bounds__(256) void moe_router_kernel(
    const float* __restrict__ x, const float* __restrict__ noise,
    const float* __restrict__ Wr, const float* __restrict__ br,
    const float* __restrict__ Wn, const float* __restrict__ bn,
    float* __restrict__ g) {
  const int wave = threadIdx.x >> 5;
  const int lane = threadIdx.x & 31;
  const int tok  = blockIdx.x * 8 + wave;
  const float* xr = x + (size_t)tok * C_DIM;

  float accR[E_NUM], accN[E_NUM];
#pragma unroll
  for (int e = 0; e < E_NUM; ++e) { accR[e] = 0.f; accN[e] = 0.f; }

  // 4096 / (32 lanes * 4 floats) = 32 iterations
  for (int it = 0; it < C_DIM / 128; ++it) {
    const int c = (it * 32 + lane) * 4;
    const float4 xv = *(const float4*)(xr + c);
#pragma unroll
    for (int e = 0; e < E_NUM; ++e) {
      const float4 wr = *(const float4*)(Wr + (size_t)e * C_DIM + c);
      const float4 wn = *(const float4*)(Wn + (size_t)e * C_DIM + c);
      accR[e] += xv.x * wr.x + xv.y * wr.y + xv.z * wr.z + xv.w * wr.w;
      accN[e] += xv.x * wn.x + xv.y * wn.y + xv.z * wn.z + xv.w * wn.w;
    }
  }

  // wave32 butterfly reduction (warpSize == 32 on gfx1250)
#pragma unroll
  for (int e = 0; e < E_NUM; ++e) {
#pragma unroll
    for (int off = 16; off > 0; off >>= 1) {
      accR[e] += __shfl_xor(accR[e], off, 32);
      accN[e] += __shfl_xor(accN[e], off, 32);
    }
  }

  // All lanes hold the full dots now; compute noisy logits redundantly.
  float noisy[E_NUM];
#pragma unroll
  for (int e = 0; e < E_NUM; ++e) {
    const float lg = accR[e] + br[e];
    const float nl = accN[e] + bn[e];
    // numerically stable softplus: max(x,0) + log1p(exp(-|x|))
    const float sp = fmaxf(nl, 0.f) + log1pf(__expf(-fabsf(nl)));
    noisy[e] = lg + noise[(size_t)tok * E_NUM + e] * sp;
  }

  // top-2 + softmax over the selected pair (others -> 0 gate)
  int i1 = 0; float v1 = noisy[0];
#pragma unroll
  for (int e = 1; e < E_NUM; ++e)
    if (noisy[e] > v1) { v1 = noisy[e]; i1 = e; }
  int i2 = -1; float v2 = -3.4e38f;
#pragma unroll
  for (int e = 0; e < E_NUM; ++e)
    if (e != i1 && noisy[e] > v2) { v2 = noisy[e]; i2 = e; }
  const float e2  = __expf(v2 - v1);     // e1 = exp(0) = 1
  const float inv = 1.f / (1.f + e2);
  if (lane < E_NUM) {
    const float gv = (lane == i1) ? inv : ((lane == i2) ? e2 * inv : 0.f);
    g[(size_t)tok * E_NUM + lane] = gv;
  }
}

// ---------------------------------------------------------------------------
// Kernel 2: gated LoRA down-projection.
//   hw[n, j] = g[n, j/64] * sum_c x[n,c] * Acat[j,c],  Acat = A as [E*R, C]
// One wave32 per 16x16 tile, v_wmma_f32_16x16x4_f32, K-loop over C in steps
// of 4. 4 waves per block share the same 16 token rows (L0 reuse of x).
// A-operand layout (32-bit 16x4): lanes 0-15 -> K {k,k+1}, lanes 16-31 ->
// K {k+2,k+3}; B-operand mirrored per output column.
// ---------------------------------------------------------------------------
__global__ __launch_bounds__(128) void moe_lora_down_kernel(
    const float* __restrict__ x, const float* __restrict__ A,
    const float* __restrict__ g, float* __restrict__ hw) {
  const int wave = threadIdx.x >> 5;
  const int lane = threadIdx.x & 31;
  const int half = lane >> 4;       // 0: K{0,1}, 1: K{2,3}
  const int m    = lane & 15;
  const int tok0 = blockIdx.x * 16;
  const int j0   = (blockIdx.y * 4 + wave) * 16;
  const int e    = j0 >> 6;         // 16-wide tile lies inside one expert

  const float* xrow = x + (size_t)(tok0 + m) * C_DIM + half * 2;
  const float* arow = A + (size_t)(j0 + m) * C_DIM + half * 2;

  v8f acc = {0.f, 0.f, 0.f, 0.f, 0.f, 0.f, 0.f, 0.f};
  for (int k = 0; k < C_DIM; k += 4) {
    const v2f av = *(const v2f*)(xrow + k);
    const v2f bv = *(const v2f*)(arow + k);
    acc = __builtin_amdgcn_wmma_f32_16x16x4_f32(
        /*neg_a=*/false, av, /*neg_b=*/false, bv,
        /*c_mod=*/(short)0, acc, /*reuse_a=*/false, /*reuse_b=*/false);
  }

  // C/D layout: VGPR v, lanes 0-15 -> M=v, lanes 16-31 -> M=v+8; N = m.
#pragma unroll
  for (int v = 0; v < 8; ++v) {
    const int row  = tok0 + v + half * 8;
    const float gate = g[(size_t)row * E_NUM + e];
    hw[(size_t)row * JT + j0 + m] = acc[v] * gate;
  }
}

// ---------------------------------------------------------------------------
// Kernel 3: LoRA up-projection.
//   out[n, o] = SCALE * sum_k hw[n,k] * W2[k,o],  W2[k,o] = B[k/64, o, k%64]
// K = 512 handled entirely by one wave -> no atomics needed.
// ---------------------------------------------------------------------------
__global__ __launch_bounds__(128) void moe_lora_up_kernel(
    const float* __restrict__ hw, const float* __restrict__ Bm,
    float* __restrict__ out) {
  const int wave = threadIdx.x >> 5;
  const int lane = threadIdx.x & 31;
  const int half = lane >> 4;
  const int m    = lane & 15;
  const int tok0 = blockIdx.x * 16;
  const int o0   = (blockIdx.y * 4 + wave) * 16;

  const float* hrow = hw + (size_t)(tok0 + m) * JT + half * 2;

  v8f acc = {0.f, 0.f, 0.f, 0.f, 0.f, 0.f, 0.f, 0.f};
  for (int k = 0; k < JT; k += 4) {
    const int e = k >> 6;           // k..k+3 never cross an expert boundary
    const int r = k & 63;
    const v2f av = *(const v2f*)(hrow + k);
    const v2f bv = *(const v2f*)(Bm + ((size_t)e * O_DIM + (o0 + m)) * R_RANK
                                 + r + half * 2);
    acc = __builtin_amdgcn_wmma_f32_16x16x4_f32(
        false, av, false, bv, (short)0, acc, false, false);
  }

#pragma unroll
  for (int v = 0; v < 8; ++v) {
    const int row = tok0 + v + half * 8;
    out[(size_t)row * O_DIM + o0 + m] = acc[v] * LORA_SCALE;
  }
}

// ---------------------------------------------------------------------------
// Host-side launcher. Workspace layout:
//   g  : N_TOK * E floats            (256 KB)
//   hw : N_TOK * (E*R) floats        (16 MB)
// ---------------------------------------------------------------------------
extern "C" void kernel_launch(void* const* d_in, const int* in_sizes, int n_in,
                              void* d_out, int out_size, void* d_ws,
                              size_t ws_size, hipStream_t stream) {
  const float* x     = (const float*)d_in[0];
  const float* noise = (const float*)d_in[1];
  const float* Wr    = (const float*)d_in[2];
  const float* br    = (const float*)d_in[3];
  const float* Wn    = (const float*)d_in[4];
  const float* bn    = (const float*)d_in[5];
  const float* A     = (const float*)d_in[6];
  const float* Bm    = (const float*)d_in[7];
  float* out = (float*)d_out;

  float* g  = (float*)d_ws;
  float* hw = g + (size_t)N_TOK * E_NUM;

  // 1) router gates
  moe_router_kernel<<<N_TOK / 8, 256, 0, stream>>>(x, noise, Wr, br, Wn, bn, g);
  // 2) gated down-projection: hw[N, 512]
  moe_lora_down_kernel<<<dim3(N_TOK / 16, JT / 64), 128, 0, stream>>>(x, A, g, hw);
  // 3) up-projection: out[N, 4096]
  moe_lora_up_kernel<<<dim3(N_TOK / 16, O_DIM / 64), 128, 0, stream>>>(hw, Bm, out);
}